// Experts_28097676051039
// MI455X (gfx1250) — compile-verified
//
#include <hip/hip_runtime.h>

typedef __attribute__((ext_vector_type(2))) float v2f;
typedef __attribute__((ext_vector_type(4))) float v4f;
typedef __attribute__((ext_vector_type(8))) float v8f;
typedef __attribute__((ext_vector_type(4))) unsigned int v4u;
typedef __attribute__((ext_vector_type(4))) int v4i;
typedef __attribute__((ext_vector_type(8))) int v8i;

namespace {
constexpr int kE    = 32;
constexpr int kN    = 1024;
constexpr int kDIn  = 3072;
constexpr int kDH   = 1024;
constexpr int kDOut = 256;

constexpr int BM    = 64;    // rows per workgroup
constexpr int KB    = 32;    // K slab depth (16 k-pairs)
constexpr int XS    = 36;    // LDS stride of X slab (32 data + 4 pad dwords, via TDM pad)
constexpr int HS    = 1028;  // LDS stride of hidden tile (4 mod 64 -> conflict-free col reads)
constexpr int WCOLS = 256;   // W slab width
constexpr int WPS   = 544;   // W slab k-pair row stride (544%64==32 -> disjoint bank halves)

constexpr int kSmemFloats = BM * HS + BM * XS + (KB / 2) * WPS;  // 65792+2304+8704 = 76800
constexpr unsigned kXsByteOff = (unsigned)(BM * HS) * 4u;        // LDS byte offset of Xs
}

static __device__ __forceinline__ v8f wmma_f32(v2f a, v2f b, v8f c) {
  // D = A(16x4 f32) * B(4x16 f32) + C(16x16 f32)
  return __builtin_amdgcn_wmma_f32_16x16x4_f32(
      /*neg_a=*/false, a, /*neg_b=*/false, b,
      /*c_mod=*/(short)0, c, /*reuse_a=*/false, /*reuse_b=*/false);
}

// ---- Tensor Data Mover: async-load one KB x BM f32 tile (row stride = row_stride
// elements) into LDS at lds_off, padding each 32-dword row by 4 dwords (-> XS=36).
static __device__ __forceinline__ void tdm_load_x_slab(const float* gsrc, unsigned lds_off,
                                                       int row_stride) {
  const unsigned long long ga = (unsigned long long)(uintptr_t)gsrc;
  v4u g0;
  g0.x = 1u;                                   // count=1 (valid user descriptor)
  g0.y = lds_off;                              // lds_addr (bytes)
  g0.z = (unsigned)(ga & 0xFFFFFFFFu);         // global_addr[31:0]
  g0.w = (unsigned)((ga >> 32) & 0x1FFFFFFu)   // global_addr[56:32]
         | (2u << 30);                         // type = 2 ("image")
  v8i g1;
  g1[0] = (2 << 16)      // data_size = 4 bytes
        | (1 << 20)      // pad_enable
        | (4 << 22)      // pad_interval: 32 dwords (0=2,1=4,2=8,3=16,4=32)
        | (3 << 25);     // pad_amount: 4 dwords (0=1,...,3=4)
  g1[1] = (KB & 0xFFFF) << 16;   // tensor_dim0[15:0] = KB (tile always in-bounds)
  g1[2] = (BM & 0xFFFF) << 16;   // tensor_dim0[31:16]=0 | tensor_dim1[15:0] = BM
  g1[3] = (KB & 0xFFFF) << 16;   // tensor_dim1[31:16]=0 | tile_dim0 = KB
  g1[4] = BM & 0xFFFF;           // tile_dim1 = BM | tile_dim2 = 0
  g1[5] = row_stride;            // tensor_dim0_stride[31:0] (data_size units)
  g1[6] = 0;                     // stride[47:32]=0 | tensor_dim1_stride[15:0]=0
  g1[7] = 0;
  const v4i z4 = {};
#if __clang_major__ >= 23
  const v8i z8 = {};
  __builtin_amdgcn_tensor_load_to_lds(g0, g1, z4, z4, z8, 0);
#else
  __builtin_amdgcn_tensor_load_to_lds(g0, g1, z4, z4, 0);
#endif
}

// ---- W slab register double-buffer: global -> regs (prefetch), regs -> LDS
// (k-pair interleaved so a B-fragment (k,k+1 | n) is one aligned 8-byte load).
static __device__ __forceinline__ void load_w_regs(v4f (&va)[4], v4f (&vb)[4],
                                                   const float* __restrict__ wsrc,
                                                   int row_stride, int tid) {
#pragma unroll
  for (int i = 0; i < 4; ++i) {
    const int idx = tid + i * 256;
    const int rp = idx >> 6;         // k-pair 0..15
    const int c4 = idx & 63;         // float4 column 0..63
    const float* s = wsrc + (size_t)(2 * rp) * row_stride + c4 * 4;
    va[i] = *(const v4f*)(s);
    vb[i] = *(const v4f*)(s + row_stride);
  }
}

static __device__ __forceinline__ void store_w_regs(float* Ws, const v4f (&va)[4],
                                                    const v4f (&vb)[4], int tid) {
#pragma unroll
  for (int i = 0; i < 4; ++i) {
    const int idx = tid + i * 256;
    const int rp = idx >> 6;
    const int c4 = idx & 63;
    v4f lo = {va[i].x, vb[i].x, va[i].y, vb[i].y};
    v4f hi = {va[i].z, vb[i].z, va[i].w, vb[i].w};
    float* d = Ws + rp * WPS + c4 * 8;
    *(v4f*)(d)     = lo;
    *(v4f*)(d + 4) = hi;
  }
}

__global__ __launch_bounds__(256, 1)
void experts_fused_wmma(const float* __restrict__ x,
                        const float* __restrict__ w1,
                        const float* __restrict__ w2,
                        float* __restrict__ out) {
  extern __shared__ float smem[];
  float* Hs = smem;                       // [BM][HS]     hidden tile (resident both GEMMs)
  float* Xs = smem + BM * HS;             // [BM][XS]     X k-slab (TDM-written)
  float* Ws = smem + BM * HS + BM * XS;   // [KB/2][WPS]  W k-slab, k-pair interleaved

  const int e    = blockIdx.y;
  const int m0   = blockIdx.x * BM;
  const int tid  = threadIdx.x;
  const int lane = tid & 31;
  const int wave = tid >> 5;
  const int wm   = wave >> 2;   // 0..1 : 32-row band
  const int wn   = wave & 3;    // 0..3 : 64-col band
  const int l16  = lane & 15;
  const int lhalf = lane >> 4;

  const float* xe  = x  + (size_t)e * kN * kDIn + (size_t)m0 * kDIn;
  const float* w1e = w1 + (size_t)e * kDIn * kDH;
  const float* w2e = w2 + (size_t)e * kDH * kDOut;
  float*       oe  = out + (size_t)e * kN * kDOut + (size_t)m0 * kDOut;

  const v8f vzero = {};

  // ================= Layer 1: Hs = relu(X @ W1) =================
  for (int ch = 0; ch < kDH / WCOLS; ++ch) {   // 4 column chunks of 256
    v8f acc[2][4];
#pragma unroll
    for (int mt = 0; mt < 2; ++mt)
#pragma unroll
      for (int nt = 0; nt < 4; ++nt) acc[mt][nt] = vzero;

    v4f wva[4], wvb[4];
    load_w_regs(wva, wvb, w1e + ch * WCOLS, kDH, tid);   // prefetch slab k0=0

    for (int k0 = 0; k0 < kDIn; k0 += KB) {
      __syncthreads();                       // previous slab fully consumed
      if (wave == 0) tdm_load_x_slab(xe + k0, kXsByteOff, kDIn);
      store_w_regs(Ws, wva, wvb, tid);       // current slab regs -> LDS
      const int k0n = (k0 + KB < kDIn) ? (k0 + KB) : k0;
      load_w_regs(wva, wvb, w1e + (size_t)k0n * kDH + ch * WCOLS, kDH, tid);  // prefetch next
      if (wave == 0) __builtin_amdgcn_s_wait_tensorcnt(0);
      __syncthreads();

#pragma unroll
      for (int kk = 0; kk < KB; kk += 4) {
        const int kl = kk + lhalf * 2;
        const int kp = (kk >> 1) + lhalf;    // k-pair row in Ws
        v2f a[2], b[4];
#pragma unroll
        for (int mt = 0; mt < 2; ++mt) {
          const int row = wm * 32 + mt * 16 + l16;
          a[mt] = *(const v2f*)(Xs + row * XS + kl);
        }
#pragma unroll
        for (int nt = 0; nt < 4; ++nt) {
          const int col = wn * 64 + nt * 16 + l16;
          b[nt] = *(const v2f*)(Ws + kp * WPS + col * 2);
        }
#pragma unroll
        for (int mt = 0; mt < 2; ++mt)
#pragma unroll
          for (int nt = 0; nt < 4; ++nt)
            acc[mt][nt] = wmma_f32(a[mt], b[nt], acc[mt][nt]);
      }
    }

    // relu + write hidden chunk into LDS (C/D layout: row = base + v + 8*lhalf, col = l16)
#pragma unroll
    for (int mt = 0; mt < 2; ++mt)
#pragma unroll
      for (int nt = 0; nt < 4; ++nt) {
        const int col = ch * WCOLS + wn * 64 + nt * 16 + l16;
        const int rbase = wm * 32 + mt * 16 + lhalf * 8;
#pragma unroll
        for (int v = 0; v < 8; ++v) {
          Hs[(rbase + v) * HS + col] = fmaxf(acc[mt][nt][v], 0.0f);
        }
      }
  }

  // ================= Layer 2: out = relu(Hs @ W2) =================
  {
    v8f acc[2][4];
#pragma unroll
    for (int mt = 0; mt < 2; ++mt)
#pragma unroll
      for (int nt = 0; nt < 4; ++nt) acc[mt][nt] = vzero;

    v4f wva[4], wvb[4];
    load_w_regs(wva, wvb, w2e, kDOut, tid);

    for (int k0 = 0; k0 < kDH; k0 += KB) {
      __syncthreads();                       // first pass also fences hidden writes
      store_w_regs(Ws, wva, wvb, tid);
      const int k0n = (k0 + KB < kDH) ? (k0 + KB) : k0;
      load_w_regs(wva, wvb, w2e + (size_t)k0n * kDOut, kDOut, tid);
      __syncthreads();

#pragma unroll
      for (int kk = 0; kk < KB; kk += 4) {
        const int kl = kk + lhalf * 2;
        const int kp = (kk >> 1) + lhalf;
        v2f a[2], b[4];
#pragma unroll
        for (int mt = 0; mt < 2; ++mt) {
          const int row = wm * 32 + mt * 16 + l16;
          a[mt] = *(const v2f*)(Hs + row * HS + k0 + kl);
        }
#pragma unroll
        for (int nt = 0; nt < 4; ++nt) {
          const int col = wn * 64 + nt * 16 + l16;
          b[nt] = *(const v2f*)(Ws + kp * WPS + col * 2);
        }
#pragma unroll
        for (int mt = 0; mt < 2; ++mt)
#pragma unroll
          for (int nt = 0; nt < 4; ++nt)
            acc[mt][nt] = wmma_f32(a[mt], b[nt], acc[mt][nt]);
      }
    }

    // relu + store to global
#pragma unroll
    for (int mt = 0; mt < 2; ++mt)
#pragma unroll
      for (int nt = 0; nt < 4; ++nt) {
        const int col = wn * 64 + nt * 16 + l16;
        const int rbase = wm * 32 + mt * 16 + lhalf * 8;
#pragma unroll
        for (int v = 0; v < 8; ++v) {
          oe[(size_t)(rbase + v) * kDOut + col] = fmaxf(acc[mt][nt][v], 0.0f);
        }
      }
  }
}

extern "C" void kernel_launch(void* const* d_in, const int* in_sizes, int n_in,
                              void* d_out, int out_size, void* d_ws, size_t ws_size,
                              hipStream_t stream) {
  (void)in_sizes; (void)n_in; (void)out_size; (void)d_ws; (void)ws_size;
  const float* x  = (const float*)d_in[0];
  const float* w1 = (const float*)d_in[1];
  const float* w2 = (const float*)d_in[2];
  float* out = (float*)d_out;

  const size_t smem = (size_t)kSmemFloats * sizeof(float);  // ~300 KB of the 320 KB WGP LDS
  (void)hipFuncSetAttribute((const void*)experts_fused_wmma,
                            hipFuncAttributeMaxDynamicSharedMemorySize, (int)smem);

  dim3 grid(kN / BM, kE);   // 16 row tiles x 32 experts
  experts_fused_wmma<<<grid, 256, smem, stream>>>(x, w1, w2, out);
}